// MyGCN_33191507264279
// MI455X (gfx1250) — compile-verified
//
#include <hip/hip_runtime.h>
#include <hip/hip_bf16.h>

// ---------------- problem constants (match reference) ----------------
#define NN      102400
#define FH      128          // F_IN == H == 128
#define EE      1638400
#define GG      256
#define KTOP    30
#define NPG     (NN / GG)    // 400 nodes per graph (batch is contiguous)
#define COUT    32
#define KER     5
#define CONVT   (KTOP - KER + 1)   // 26
#define FLATF   (COUT * CONVT)     // 832
#define BNEPS   1e-5f

typedef __attribute__((ext_vector_type(2))) float v2f;
typedef __attribute__((ext_vector_type(8))) float v8f;

// ---------------- utility ----------------
__global__ void zero_f32_k(float* __restrict__ p, long n) {
    long i = (long)blockIdx.x * blockDim.x + threadIdx.x;
    if (i < n) p[i] = 0.0f;
}

// ---------------- degree: cnt[dst] += 1 ----------------
__global__ void degree_k(const int* __restrict__ dst, float* __restrict__ deg) {
    int e = blockIdx.x * blockDim.x + threadIdx.x;
    if (e < EE) atomicAdd(&deg[dst[e]], 1.0f);
}

// ---------------- edge scatter-add: agg[dst] += feat[src] ----------------
// one wave32 per edge, 4 floats per lane (float4 load, 4 f32 atomics)
__global__ void scatter_add_k(const float* __restrict__ feat,
                              const int* __restrict__ src,
                              const int* __restrict__ dst,
                              float* __restrict__ agg) {
    long tid  = (long)blockIdx.x * blockDim.x + threadIdx.x;
    int  e    = (int)(tid >> 5);
    int  lane = (int)(tid & 31);
    if (e >= EE) return;
    int s = src[e], d = dst[e];
    const float4 v = *(const float4*)(feat + (size_t)s * FH + lane * 4);
    float* a = agg + (size_t)d * FH + lane * 4;
    atomicAdd(a + 0, v.x);
    atomicAdd(a + 1, v.y);
    atomicAdd(a + 2, v.z);
    atomicAdd(a + 3, v.w);
}

// ---------------- SAGE GEMM via V_WMMA_F32_16X16X4_F32 ----------------
// OUT[row, col] = (AGG[row]/max(deg,1)) @ Wl + bl + X[row] @ Wr
// grid.x = NN/16 (row tiles), 256 threads = 8 waves, wave w -> col tile w.
// A fragment (16x4 f32, 2 VGPRs/lane): lane group g (lane>>4) holds K = 2g+{0,1}.
// B fragment (4x16 f32): lane group g holds rows K = 2g+{0,1}, N = lane&15.
// C/D (16x16 f32, 8 VGPRs): element i -> M = i + 8g, N = lane&15.
__global__ __launch_bounds__(256) void sage_gemm_k(
        const float* __restrict__ X, const float* __restrict__ AGG,
        const float* __restrict__ DEG,
        const float* __restrict__ Wl, const float* __restrict__ bl,
        const float* __restrict__ Wr, float* __restrict__ OUT) {
    const int wave = threadIdx.x >> 5;
    const int lane = threadIdx.x & 31;
    const int grp  = lane >> 4;          // 0 or 1
    const int l16  = lane & 15;
    const int row0 = blockIdx.x << 4;
    const int col  = (wave << 4) + l16;  // output column for this lane
    const int arow = row0 + l16;         // A-matrix row held by this lane

    const float inv = 1.0f / fmaxf(DEG[arow], 1.0f);
    const float* __restrict__ xr = X   + (size_t)arow * FH;
    const float* __restrict__ ar = AGG + (size_t)arow * FH;

    v8f acc = {};
    // term: X @ Wr
    #pragma unroll 4
    for (int k = 0; k < FH; k += 4) {
        const int kk = k + (grp << 1);
        v2f a, b;
        a[0] = xr[kk];
        a[1] = xr[kk + 1];
        b[0] = Wr[(size_t)kk * FH + col];
        b[1] = Wr[(size_t)(kk + 1) * FH + col];
        acc = __builtin_amdgcn_wmma_f32_16x16x4_f32(false, a, false, b,
                                                    (short)0, acc, false, false);
    }
    // term: mean @ Wl  (mean = agg * inv, folded into A fragment load)
    #pragma unroll 4
    for (int k = 0; k < FH; k += 4) {
        const int kk = k + (grp << 1);
        v2f a, b;
        a[0] = ar[kk] * inv;
        a[1] = ar[kk + 1] * inv;
        b[0] = Wl[(size_t)kk * FH + col];
        b[1] = Wl[(size_t)(kk + 1) * FH + col];
        acc = __builtin_amdgcn_wmma_f32_16x16x4_f32(false, a, false, b,
                                                    (short)0, acc, false, false);
    }
    const float bias = bl[col];
    #pragma unroll
    for (int i = 0; i < 8; ++i) {
        const int m = i + (grp << 3);
        OUT[(size_t)(row0 + m) * FH + col] = acc[i] + bias;
    }
}

// ---------------- BatchNorm over N rows ----------------
#define BN_ROWS 256
__global__ void bn_stats_k(const float* __restrict__ h, float* __restrict__ stats) {
    const int col = threadIdx.x;                 // 128 threads
    const long r0 = (long)blockIdx.x * BN_ROWS;
    float s = 0.0f, s2 = 0.0f;
    for (int r = 0; r < BN_ROWS; ++r) {
        float v = h[(r0 + r) * FH + col];
        s += v; s2 += v * v;
    }
    atomicAdd(&stats[col], s);
    atomicAdd(&stats[FH + col], s2);
}

__global__ void bn_finalize_k(const float* __restrict__ stats,
                              const float* __restrict__ gam,
                              const float* __restrict__ bet,
                              float* __restrict__ scsh) {
    const int c = threadIdx.x;
    const float invN = 1.0f / (float)NN;
    float mu   = stats[c] * invN;
    float var  = stats[FH + c] * invN - mu * mu;
    float rstd = rsqrtf(var + BNEPS);
    float sc   = rstd * gam[c];
    scsh[c]      = sc;
    scsh[FH + c] = bet[c] - mu * sc;
}

__global__ void bn_apply_relu_k(float* __restrict__ h,
                                const float* __restrict__ scsh, long n) {
    long i = (long)blockIdx.x * blockDim.x + threadIdx.x;
    if (i >= n) return;
    int c = (int)(i & (FH - 1));
    h[i] = fmaxf(h[i] * scsh[c] + scsh[FH + c], 0.0f);
}

// ---------------- per-graph descending top-K on feature 127 ----------------
// stable-sort semantics: ties broken by lower node index
__global__ __launch_bounds__(512) void topk_k(const float* __restrict__ h,
                                              int* __restrict__ outIdx) {
    __shared__ float v[NPG];
    __shared__ float rv[512];
    __shared__ int   ri[512];
    const int g = blockIdx.x;
    const int t = threadIdx.x;
    if (t < NPG) v[t] = h[(size_t)(g * NPG + t) * FH + (FH - 1)];
    __syncthreads();
    for (int k = 0; k < KTOP; ++k) {
        rv[t] = (t < NPG) ? v[t] : -3.4e38f;
        ri[t] = (t < NPG) ? t : 0x7fffffff;
        __syncthreads();
        for (int s = 256; s > 0; s >>= 1) {
            if (t < s) {
                float ov = rv[t + s]; int oi = ri[t + s];
                if (ov > rv[t] || (ov == rv[t] && oi < ri[t])) { rv[t] = ov; ri[t] = oi; }
            }
            __syncthreads();
        }
        if (t == 0) {
            outIdx[g * KTOP + k] = g * NPG + ri[0];
            v[ri[0]] = -3.4e38f;
        }
        __syncthreads();
    }
}

// ---------------- fused head: gather -> conv1d -> relu -> lin1 -> relu -> lin2 ----------------
__global__ __launch_bounds__(128) void head_k(
        const float* __restrict__ h, const int* __restrict__ idx,
        const float* __restrict__ convW, const float* __restrict__ convB,
        const float* __restrict__ lin1W, const float* __restrict__ lin1b,
        const float* __restrict__ lin2W, const float* __restrict__ lin2b,
        float* __restrict__ out) {
    __shared__ float p[KTOP][FH];   // gathered rows (pre-transpose view)
    __shared__ float c[FLATF];      // conv output, flattened m = o*26 + t
    __shared__ float f[FH];
    const int g = blockIdx.x;
    const int t = threadIdx.x;      // 128 threads

    for (int e = t; e < KTOP * FH; e += 128) {
        int kk = e >> 7, col = e & (FH - 1);
        p[kk][col] = h[(size_t)idx[g * KTOP + kk] * FH + col];
    }
    __syncthreads();

    // conv: c[o][tt] = relu(convB[o] + sum_i sum_j p[tt+j][i] * convW[o][i][j])
    for (int e = t; e < FLATF; e += 128) {
        int o = e / CONVT, tt = e % CONVT;
        float s = convB[o];
        for (int i = 0; i < FH; ++i) {
            const float* w = &convW[((size_t)o * FH + i) * KER];
            #pragma unroll
            for (int j = 0; j < KER; ++j) s += p[tt + j][i] * w[j];
        }
        c[e] = fmaxf(s, 0.0f);
    }
    __syncthreads();

    // lin1: 832 -> 128
    {
        float s = lin1b[t];
        for (int m = 0; m < FLATF; ++m) s += c[m] * lin1W[(size_t)m * FH + t];
        f[t] = fmaxf(s, 0.0f);
    }
    __syncthreads();

    // lin2: 128 -> 2
    if (t < 2) {
        float s = lin2b[t];
        for (int hh = 0; hh < FH; ++hh) s += f[hh] * lin2W[hh * 2 + t];
        out[g * 2 + t] = s;
    }
}

// ---------------- host orchestration ----------------
extern "C" void kernel_launch(void* const* d_in, const int* in_sizes, int n_in,
                              void* d_out, int out_size, void* d_ws, size_t ws_size,
                              hipStream_t stream) {
    const float* x     = (const float*)d_in[0];
    const int*   edge  = (const int*)d_in[1];
    const int*   src   = edge;          // edge_index[0]
    const int*   dst   = edge + EE;     // edge_index[1]
    const float* W1l   = (const float*)d_in[3];
    const float* b1l   = (const float*)d_in[4];
    const float* W1r   = (const float*)d_in[5];
    const float* W2l   = (const float*)d_in[6];
    const float* b2l   = (const float*)d_in[7];
    const float* W2r   = (const float*)d_in[8];
    const float* g1    = (const float*)d_in[9];
    const float* be1   = (const float*)d_in[10];
    const float* g2    = (const float*)d_in[11];
    const float* be2   = (const float*)d_in[12];
    const float* convW = (const float*)d_in[13];
    const float* convB = (const float*)d_in[14];
    const float* lin1W = (const float*)d_in[15];
    const float* lin1b = (const float*)d_in[16];
    const float* lin2W = (const float*)d_in[17];
    const float* lin2b = (const float*)d_in[18];
    float*       out   = (float*)d_out;

    // workspace layout (floats)
    const size_t NH = (size_t)NN * FH;
    float* ws    = (float*)d_ws;
    float* agg   = ws;                 // N*H
    float* hA    = agg   + NH;         // N*H (layer1 pre/post, in place)
    float* hB    = hA    + NH;         // N*H (layer2 pre/post, in place)
    float* deg   = hB    + NH;         // N
    float* stats = deg   + NN;         // 2*H (sum, sumsq)
    float* scsh  = stats + 2 * FH;     // 2*H (scale, shift)
    int*   topk  = (int*)(scsh + 2 * FH); // G*K

    const int TB = 256;
    const long nNH = (long)NH;

    // ---- layer 1 ----
    zero_f32_k<<<(int)((nNH + TB - 1) / TB), TB, 0, stream>>>(agg, nNH);
    zero_f32_k<<<(NN + TB - 1) / TB, TB, 0, stream>>>(deg, NN);
    zero_f32_k<<<1, TB, 0, stream>>>(stats, 2 * FH);
    degree_k<<<(EE + TB - 1) / TB, TB, 0, stream>>>(dst, deg);
    scatter_add_k<<<(int)(((long)EE * 32) / TB), TB, 0, stream>>>(x, src, dst, agg);
    sage_gemm_k<<<NN / 16, 256, 0, stream>>>(x, agg, deg, W1l, b1l, W1r, hA);
    bn_stats_k<<<NN / BN_ROWS, FH, 0, stream>>>(hA, stats);
    bn_finalize_k<<<1, FH, 0, stream>>>(stats, g1, be1, scsh);
    bn_apply_relu_k<<<(int)((nNH + TB - 1) / TB), TB, 0, stream>>>(hA, scsh, nNH);

    // ---- layer 2 ----
    zero_f32_k<<<(int)((nNH + TB - 1) / TB), TB, 0, stream>>>(agg, nNH);
    zero_f32_k<<<1, TB, 0, stream>>>(stats, 2 * FH);
    scatter_add_k<<<(int)(((long)EE * 32) / TB), TB, 0, stream>>>(hA, src, dst, agg);
    sage_gemm_k<<<NN / 16, 256, 0, stream>>>(hA, agg, deg, W2l, b2l, W2r, hB);
    bn_stats_k<<<NN / BN_ROWS, FH, 0, stream>>>(hB, stats);
    bn_finalize_k<<<1, FH, 0, stream>>>(stats, g2, be2, scsh);
    bn_apply_relu_k<<<(int)((nNH + TB - 1) / TB), TB, 0, stream>>>(hB, scsh, nNH);

    // ---- sort-pool + head ----
    topk_k<<<GG, 512, 0, stream>>>(hB, topk);
    head_k<<<GG, 128, 0, stream>>>(hB, topk, convW, convB,
                                   lin1W, lin1b, lin2W, lin2b, out);
}